// GraphConvolution_38371237822945
// MI455X (gfx1250) — compile-verified
//
#include <hip/hip_runtime.h>
#include <math.h>

#define D 128
#define EPS 1e-9f

typedef __attribute__((ext_vector_type(2))) float v2f;
typedef __attribute__((ext_vector_type(8))) float v8f;

// ---------------------------------------------------------------------------
// Kernel 1: SpMM scatter.  One wave32 per edge; lane l owns 4 consecutive
// features (float4).  feat[row[e]][:] += val[e] * x[col[e]][:]
// x (51.2MB) and feat (51.2MB) are both L2-resident on MI455X (192MB L2),
// so the random gathers and hardware f32 atomics run at L2 bandwidth.
// ---------------------------------------------------------------------------
__global__ void __launch_bounds__(256) spmm_kernel(
    const float* __restrict__ x, const int* __restrict__ erow,
    const int* __restrict__ ecol, const float* __restrict__ eval,
    float* __restrict__ feat, int n_edges)
{
  const int t = blockIdx.x * blockDim.x + threadIdx.x;
  const int e = t >> 5;
  const int lane = t & 31;
  if (e >= n_edges) return;

  const int r = erow[e];
  const int c = ecol[e];
  const float v = eval[e];

  const float4 xv = *(const float4*)(x + (long)c * D + lane * 4);
  float* dst = feat + (long)r * D + lane * 4;
  unsafeAtomicAdd(dst + 0, xv.x * v);   // global_atomic_add_f32
  unsafeAtomicAdd(dst + 1, xv.y * v);
  unsafeAtomicAdd(dst + 2, xv.z * v);
  unsafeAtomicAdd(dst + 3, xv.w * v);
}

// ---------------------------------------------------------------------------
// Kernel 2: fused (feat @ W^T + b) -> ELU -> per-row mean/var norm.
// 256 threads = 8 waves per block; block owns 16 node rows.
// Wave w computes the 16x16 output tile for columns [16w, 16w+16) with
// V_WMMA_F32_16X16X4_F32 (full fp32, 32 K-steps of 4).
// ---------------------------------------------------------------------------
__global__ void __launch_bounds__(256) gemm_elu_norm_kernel(
    const float* __restrict__ feat, const float* __restrict__ W,
    const float* __restrict__ bias, const float* __restrict__ scale,
    const float* __restrict__ offset, float* __restrict__ out, int n_nodes)
{
  __shared__ float sA[16][132];   // feat tile, padded (stride 132 -> bank stride 4)
  __shared__ float sO[16][132];   // post-ELU tile for the row reduction

  const int tid  = threadIdx.x;
  const int wave = tid >> 5;
  const int lane = tid & 31;
  const int row0 = blockIdx.x * 16;

  // Stage 16x128 feat tile into LDS (512 float4 loads, 2 per thread).
  #pragma unroll
  for (int i = 0; i < 2; ++i) {
    const int e4 = tid * 2 + i;         // 0..511
    const int rr = e4 >> 5;             // row within tile
    const int cc = (e4 & 31) * 4;       // col within row
    float4 v = make_float4(0.f, 0.f, 0.f, 0.f);
    if (row0 + rr < n_nodes)
      v = *(const float4*)(feat + (long)(row0 + rr) * D + cc);
    *(float4*)&sA[rr][cc] = v;
  }
  __syncthreads();

  // WMMA operand indexing (ISA 16x4 f32 A layout):
  //   lanes 0-15 : M=lane,    VGPR0=K+0, VGPR1=K+1
  //   lanes 16-31: M=lane-16, VGPR0=K+2, VGPR1=K+3
  // B (4x16) mirrors with N across lanes; B[k][n] = W[n][k].
  const int col0 = wave * 16;
  const int mn   = lane & 15;          // M for A, N for B
  const int kh   = (lane >> 4) * 2;    // K sub-offset for this lane half
  const float* wrow = W + (long)(col0 + mn) * D;
  __builtin_prefetch(wrow, 0, 0);      // global_prefetch_b8 (W is 64KB, L2-hot)

  v8f acc = {};
  #pragma unroll
  for (int k = 0; k < D; k += 4) {
    v2f a, b2;
    a.x  = sA[mn][k + kh];
    a.y  = sA[mn][k + kh + 1];
    b2.x = wrow[k + kh];
    b2.y = wrow[k + kh + 1];
    // (neg_a, A, neg_b, B, c_mod, C, reuse_a, reuse_b)
    acc = __builtin_amdgcn_wmma_f32_16x16x4_f32(false, a, false, b2,
                                                (short)0, acc, false, false);
  }

  // C/D layout: VGPR v holds (M = v + 8*(lane>=16), N = lane&15).
  // Apply bias + ELU, park tile in LDS for the cross-lane row reduction.
  const float bv = bias[col0 + mn];
  #pragma unroll
  for (int v = 0; v < 8; ++v) {
    const int mm = v + 8 * (lane >> 4);
    float val = acc[v] + bv;
    val = (val > 0.f) ? val : (expf(val) - 1.f);   // ELU
    sO[mm][col0 + mn] = val;
  }
  __syncthreads();

  // Per-row normalization: wave w handles rows 2w and 2w+1.
  // 128 values/row = 4 per lane; reduce sum & sumsq with a shfl_xor tree.
  const float4 sc = *(const float4*)(scale  + lane * 4);
  const float4 of = *(const float4*)(offset + lane * 4);
  #pragma unroll
  for (int ri = 0; ri < 2; ++ri) {
    const int r = wave * 2 + ri;
    const float4 vals = *(const float4*)&sO[r][lane * 4];
    float s  = vals.x + vals.y + vals.z + vals.w;
    float s2 = vals.x * vals.x + vals.y * vals.y +
               vals.z * vals.z + vals.w * vals.w;
    #pragma unroll
    for (int m = 16; m; m >>= 1) {
      s  += __shfl_xor(s,  m, 32);
      s2 += __shfl_xor(s2, m, 32);
    }
    const float mean = s * (1.0f / 128.0f);
    const float var  = s2 * (1.0f / 128.0f) - mean * mean + EPS;  // biased var
    const float rinv = rsqrtf(var);
    float4 o;
    o.x = (vals.x - mean) * sc.x * rinv + of.x;
    o.y = (vals.y - mean) * sc.y * rinv + of.y;
    o.z = (vals.z - mean) * sc.z * rinv + of.z;
    o.w = (vals.w - mean) * sc.w * rinv + of.w;
    if (row0 + r < n_nodes)
      *(float4*)(out + (long)(row0 + r) * D + lane * 4) = o;
  }
}

// ---------------------------------------------------------------------------
extern "C" void kernel_launch(void* const* d_in, const int* in_sizes, int n_in,
                              void* d_out, int out_size, void* d_ws, size_t ws_size,
                              hipStream_t stream) {
  const float* x      = (const float*)d_in[0];
  const int*   erow   = (const int*)  d_in[1];
  const int*   ecol   = (const int*)  d_in[2];
  const float* eval   = (const float*)d_in[3];
  const float* W      = (const float*)d_in[4];
  const float* b      = (const float*)d_in[5];
  const float* scale  = (const float*)d_in[6];
  const float* offset = (const float*)d_in[7];
  float*       out    = (float*)d_out;

  const int n_nodes = in_sizes[0] / D;
  const int n_edges = in_sizes[1];

  float* feat = (float*)d_ws;                       // [n_nodes, 128] scratch
  const size_t feat_bytes = (size_t)n_nodes * D * sizeof(float);
  hipMemsetAsync(feat, 0, feat_bytes, stream);      // graph-capturable

  const long spmm_threads = (long)n_edges * 32;
  const int  spmm_blocks  = (int)((spmm_threads + 255) / 256);
  spmm_kernel<<<spmm_blocks, 256, 0, stream>>>(x, erow, ecol, eval, feat, n_edges);

  const int tiles = (n_nodes + 15) / 16;
  gemm_elu_norm_kernel<<<tiles, 256, 0, stream>>>(feat, W, b, scale, offset, out, n_nodes);
}